// _LightGCN_45079976739440
// MI455X (gfx1250) — compile-verified
//
#include <hip/hip_runtime.h>
#include <hip/hip_bf16.h>

typedef __attribute__((ext_vector_type(2))) float v2f;
typedef __attribute__((ext_vector_type(8))) float v8f;

#define U_NODES 50000
#define I_NODES 50000
#define N_NODES 100000
#define DIM     64
#define HID     128
#define EDGES   1600000
#define BATCH   1024

// ---------------- WMMA helpers (V_WMMA_F32_16X16X4_F32, chained over K) ----
__device__ __forceinline__ v8f wmma_step(v2f a, v2f b, v8f c) {
  return __builtin_amdgcn_wmma_f32_16x16x4_f32(false, a, false, b, (short)0, c,
                                               false, false);
}

// C(16x16) += A(16xK row-major, lda) * B(K x 16 row-major, ldb)
__device__ __forceinline__ v8f wmma_AB(const float* __restrict__ A, int lda,
                                       const float* __restrict__ B, int ldb,
                                       int K, v8f c, int lane) {
  const int mn = lane & 15;
  const int kh = (lane >> 4) << 1;  // 0 or 2
  for (int k0 = 0; k0 < K; k0 += 4) {
    v2f a, b;
    a.x = A[mn * lda + k0 + kh];
    a.y = A[mn * lda + k0 + kh + 1];
    b.x = B[(k0 + kh) * ldb + mn];
    b.y = B[(k0 + kh + 1) * ldb + mn];
    c = wmma_step(a, b, c);
  }
  return c;
}

// C(16x16) += A(16xK, lda) * Bt(16 x K, ldb)^T   (B[k][n] = Bt[n*ldb+k])
__device__ __forceinline__ v8f wmma_ABt(const float* __restrict__ A, int lda,
                                        const float* __restrict__ Bt, int ldb,
                                        int K, v8f c, int lane) {
  const int mn = lane & 15;
  const int kh = (lane >> 4) << 1;
  for (int k0 = 0; k0 < K; k0 += 4) {
    v2f a, b;
    a.x = A[mn * lda + k0 + kh];
    a.y = A[mn * lda + k0 + kh + 1];
    b.x = Bt[mn * ldb + k0 + kh];
    b.y = Bt[mn * ldb + k0 + kh + 1];
    c = wmma_step(a, b, c);
  }
  return c;
}

__device__ __forceinline__ float wave_sum(float v) {
  for (int o = 16; o > 0; o >>= 1) v += __shfl_xor(v, o, 32);
  return v;
}

// ---------------- elementwise kernels --------------------------------------
__global__ void k_init(const float* __restrict__ ue, const float* __restrict__ ie,
                       float* __restrict__ ego, float* __restrict__ acc) {
  int i = blockIdx.x * 256 + threadIdx.x;            // exact: N*DIM threads
  float v = (i < U_NODES * DIM) ? ue[i] : ie[i - U_NODES * DIM];
  ego[i] = v;
  acc[i] = v;
}

__global__ void k_zero(float* __restrict__ p) {
  p[blockIdx.x * 256 + threadIdx.x] = 0.0f;
}

__global__ void k_addcopy(float* __restrict__ acc, const float* __restrict__ nxt,
                          float* __restrict__ ego) {
  int i = blockIdx.x * 256 + threadIdx.x;
  float v = nxt[i];
  acc[i] += v;
  ego[i] = v;
}

// X = acc/4 + 0.1*noise  (noise table chosen per row)
__global__ void k_noise(const float* __restrict__ acc,
                        const float* __restrict__ nu, const float* __restrict__ ni,
                        float* __restrict__ X) {
  int i = blockIdx.x * 256 + threadIdx.x;
  int row = i >> 6;
  float nz = (row < U_NODES) ? nu[i] : ni[i - U_NODES * DIM];
  X[i] = acc[i] * 0.25f + 0.1f * nz;
}

// ---------------- SPMM: y[rows[e]] += vals[e]*x[cols[e]] (one wave / edge) --
__global__ void k_spmm(const int* __restrict__ rows, const int* __restrict__ cols,
                       const float* __restrict__ vals, const float* __restrict__ x,
                       float* __restrict__ y) {
  unsigned t = blockIdx.x * 256 + threadIdx.x;       // exact: EDGES*32 threads
  unsigned e = t >> 5;
  int lane = t & 31;
  int r = rows[e], c = cols[e];
  float v = vals[e];
  const float2 xv = *(const float2*)(x + (size_t)c * DIM + lane * 2);
  float* out = y + (size_t)r * DIM + lane * 2;
  unsafeAtomicAdd(out, v * xv.x);
  unsafeAtomicAdd(out + 1, v * xv.y);
}

// ---------------- L2 row normalize of acc/4 -> E (one wave / row) ----------
__global__ void k_norm(const float* __restrict__ acc, float* __restrict__ E) {
  int row = blockIdx.x * 8 + (threadIdx.x >> 5);     // exact: N waves
  int lane = threadIdx.x & 31;
  float2 x = *(const float2*)(acc + (size_t)row * DIM + lane * 2);
  x.x *= 0.25f; x.y *= 0.25f;
  float ss = wave_sum(x.x * x.x + x.y * x.y);
  float inv = 1.0f / fmaxf(sqrtf(ss), 1e-12f);
  float2 o; o.x = x.x * inv; o.y = x.y * inv;
  *(float2*)(E + (size_t)row * DIM + lane * 2) = o;
}

// ---------------- denoiser MLP layer 1: H = relu(X@W1 + b1) ---------------
__global__ __launch_bounds__(256) void k_mlp1(
    const float* __restrict__ X,
    const float* __restrict__ duW1, const float* __restrict__ dub1,
    const float* __restrict__ diW1, const float* __restrict__ dib1,
    float* __restrict__ H) {
  int wave = blockIdx.x * 8 + (threadIdx.x >> 5);    // (N/16)*(HID/16) waves
  int lane = threadIdx.x & 31;
  int mt = wave >> 3, nt = wave & 7;
  int m0 = mt * 16, n0 = nt * 16;
  bool usr = m0 < U_NODES;                           // tiles never straddle
  const float* W = usr ? duW1 : diW1;
  const float* bb = usr ? dub1 : dib1;
  v8f c = {0.f, 0.f, 0.f, 0.f, 0.f, 0.f, 0.f, 0.f};
  c = wmma_AB(X + (size_t)m0 * DIM, DIM, W + n0, HID, DIM, c, lane);
  int hi = lane >> 4, n = lane & 15;
  float bias = bb[n0 + n];
  for (int r = 0; r < 8; r++) {
    int m = m0 + r + 8 * hi;
    float v = c[r] + bias;
    H[(size_t)m * HID + n0 + n] = v > 0.f ? v : 0.f;
  }
}

// ---------------- denoiser MLP layer 2: Z = H@W2 + b2 ----------------------
__global__ __launch_bounds__(256) void k_mlp2(
    const float* __restrict__ H,
    const float* __restrict__ duW2, const float* __restrict__ dub2,
    const float* __restrict__ diW2, const float* __restrict__ dib2,
    float* __restrict__ Z) {
  int wave = blockIdx.x * 8 + (threadIdx.x >> 5);    // (N/16)*(DIM/16) waves
  int lane = threadIdx.x & 31;
  int mt = wave >> 2, nt = wave & 3;
  int m0 = mt * 16, n0 = nt * 16;
  bool usr = m0 < U_NODES;
  const float* W = usr ? duW2 : diW2;
  const float* bb = usr ? dub2 : dib2;
  v8f c = {0.f, 0.f, 0.f, 0.f, 0.f, 0.f, 0.f, 0.f};
  c = wmma_AB(H + (size_t)m0 * HID, HID, W + n0, DIM, HID, c, lane);
  int hi = lane >> 4, n = lane & 15;
  float bias = bb[n0 + n];
  for (int r = 0; r < 8; r++) {
    int m = m0 + r + 8 * hi;
    Z[(size_t)m * DIM + n0 + n] = c[r] + bias;
  }
}

// ------- gather batch rows of E1 + positive dots (one wave / batch elem) ---
__global__ void k_gather(const int* __restrict__ users, const int* __restrict__ items,
                         const float* __restrict__ E1, const float* __restrict__ E2,
                         float* __restrict__ U1g, float* __restrict__ I1g,
                         float* __restrict__ posu, float* __restrict__ posi) {
  int b = blockIdx.x * 8 + (threadIdx.x >> 5);
  int lane = threadIdx.x & 31;
  int u = users[b];
  int itr = U_NODES + items[b];
  float2 e1u = *(const float2*)(E1 + (size_t)u * DIM + lane * 2);
  float2 e2u = *(const float2*)(E2 + (size_t)u * DIM + lane * 2);
  float2 e1i = *(const float2*)(E1 + (size_t)itr * DIM + lane * 2);
  float2 e2i = *(const float2*)(E2 + (size_t)itr * DIM + lane * 2);
  *(float2*)(U1g + (size_t)b * DIM + lane * 2) = e1u;
  *(float2*)(I1g + (size_t)b * DIM + lane * 2) = e1i;
  float pu = wave_sum(e1u.x * e2u.x + e1u.y * e2u.y);
  float pi = wave_sum(e1i.x * e2i.x + e1i.y * e2i.y);
  if (lane == 0) { posu[b] = pu; posi[b] = pi; }
}

// ------- sup_logits[b] = z_u . z_i - z_u . z_ni ----------------------------
__global__ void k_sup(const int* __restrict__ users, const int* __restrict__ items,
                      const int* __restrict__ negs, const float* __restrict__ Z,
                      float* __restrict__ out) {
  int b = blockIdx.x * 8 + (threadIdx.x >> 5);
  int lane = threadIdx.x & 31;
  int u = users[b];
  int it = U_NODES + items[b];
  int ni = U_NODES + negs[b];
  float2 zu = *(const float2*)(Z + (size_t)u * DIM + lane * 2);
  float2 zi = *(const float2*)(Z + (size_t)it * DIM + lane * 2);
  float2 zn = *(const float2*)(Z + (size_t)ni * DIM + lane * 2);
  float s = wave_sum(zu.x * (zi.x - zn.x) + zu.y * (zi.y - zn.y));
  if (lane == 0) out[b] = s;
}

// ------- SSL logits: out[b, n] = Ag[b] . E2[n0+n] - pos[b]  (store-bound) --
__global__ __launch_bounds__(256) void k_ssl(
    const float* __restrict__ Ag, const float* __restrict__ E2part,
    const float* __restrict__ pos, float* __restrict__ out, int ldo) {
  int wave = blockIdx.x * 8 + (threadIdx.x >> 5);    // (BATCH/16)*(50000/16)
  int lane = threadIdx.x & 31;
  int mt = wave / 3125, nt = wave % 3125;
  int m0 = mt * 16, n0 = nt * 16;
  v8f c = {0.f, 0.f, 0.f, 0.f, 0.f, 0.f, 0.f, 0.f};
  c = wmma_ABt(Ag + (size_t)m0 * DIM, DIM, E2part + (size_t)n0 * DIM, DIM,
               DIM, c, lane);
  int hi = lane >> 4, n = lane & 15;
  for (int r = 0; r < 8; r++) {
    int m = m0 + r + 8 * hi;
    float v = c[r] - pos[m];
    __builtin_nontemporal_store(v, out + (size_t)m * ldo + n0 + n);
  }
}

// ---------------------------------------------------------------------------
extern "C" void kernel_launch(void* const* d_in, const int* in_sizes, int n_in,
                              void* d_out, int out_size, void* d_ws, size_t ws_size,
                              hipStream_t stream) {
  const int*   adj_rows  = (const int*)d_in[0];
  const int*   adj_cols  = (const int*)d_in[1];
  const float* adj_vals  = (const float*)d_in[2];
  const int*   s1_rows   = (const int*)d_in[3];
  const int*   s1_cols   = (const int*)d_in[4];
  const float* s1_vals   = (const float*)d_in[5];
  const int*   s2_rows   = (const int*)d_in[6];
  const int*   s2_cols   = (const int*)d_in[7];
  const float* s2_vals   = (const float*)d_in[8];
  const int*   users     = (const int*)d_in[9];
  const int*   items     = (const int*)d_in[10];
  const int*   negs      = (const int*)d_in[11];
  const float* user_emb  = (const float*)d_in[12];
  const float* item_emb  = (const float*)d_in[13];
  const float* du_W1     = (const float*)d_in[14];
  const float* du_b1     = (const float*)d_in[15];
  const float* du_W2     = (const float*)d_in[16];
  const float* du_b2     = (const float*)d_in[17];
  const float* di_W1     = (const float*)d_in[18];
  const float* di_b1     = (const float*)d_in[19];
  const float* di_W2     = (const float*)d_in[20];
  const float* di_b2     = (const float*)d_in[21];
  const float* noise_u   = (const float*)d_in[22];
  const float* noise_i   = (const float*)d_in[23];
  float* out = (float*)d_out;

  const size_t ND = (size_t)N_NODES * DIM;
  float* ws   = (float*)d_ws;
  float* ego  = ws;
  float* nxt  = ego + ND;
  float* acc  = nxt + ND;
  float* Z    = acc + ND;
  float* E1   = Z + ND;
  float* E2   = E1 + ND;
  float* H    = E2 + ND;
  float* U1g  = H + (size_t)N_NODES * HID;
  float* I1g  = U1g + (size_t)BATCH * DIM;
  float* posu = I1g + (size_t)BATCH * DIM;
  float* posi = posu + BATCH;

  const int GB_ND   = (N_NODES * DIM) / 256;  // 25000 blocks, exact
  const int GB_SPMM = (EDGES * 32) / 256;     // 200000 blocks, exact

  auto run_gcn = [&](const int* r, const int* c, const float* v) {
    k_init<<<GB_ND, 256, 0, stream>>>(user_emb, item_emb, ego, acc);
    for (int l = 0; l < 3; l++) {
      k_zero<<<GB_ND, 256, 0, stream>>>(nxt);
      k_spmm<<<GB_SPMM, 256, 0, stream>>>(r, c, v, ego, nxt);
      k_addcopy<<<GB_ND, 256, 0, stream>>>(acc, nxt, ego);
    }
  };

  // Main graph -> denoised Z
  run_gcn(adj_rows, adj_cols, adj_vals);
  k_noise<<<GB_ND, 256, 0, stream>>>(acc, noise_u, noise_i, ego);  // reuse ego as X
  k_mlp1<<<(N_NODES / 16) * (HID / 16) / 8, 256, 0, stream>>>(ego, du_W1, du_b1,
                                                              di_W1, di_b1, H);
  k_mlp2<<<(N_NODES / 16) * (DIM / 16) / 8, 256, 0, stream>>>(H, du_W2, du_b2,
                                                              di_W2, di_b2, Z);
  // Subgraph views -> normalized E1, E2
  run_gcn(s1_rows, s1_cols, s1_vals);
  k_norm<<<N_NODES / 8, 256, 0, stream>>>(acc, E1);
  run_gcn(s2_rows, s2_cols, s2_vals);
  k_norm<<<N_NODES / 8, 256, 0, stream>>>(acc, E2);

  // Batch gathers + positives + sup logits
  k_gather<<<BATCH / 8, 256, 0, stream>>>(users, items, E1, E2, U1g, I1g, posu, posi);
  k_sup<<<BATCH / 8, 256, 0, stream>>>(users, items, negs, Z, out);

  // SSL logits (store-bound GEMMs, non-temporal output)
  const int GB_SSL = (BATCH / 16) * (U_NODES / 16) / 8;  // 25000 blocks
  k_ssl<<<GB_SSL, 256, 0, stream>>>(U1g, E2, posu, out + BATCH, U_NODES);
  k_ssl<<<GB_SSL, 256, 0, stream>>>(I1g, E2 + (size_t)U_NODES * DIM, posi,
                                    out + BATCH + (size_t)BATCH * U_NODES, I_NODES);
}